// L2ContrastiveLoss_19155554140556
// MI455X (gfx1250) — compile-verified
//
#include <hip/hip_runtime.h>
#include <hip/hip_bf16.h>

typedef float v2f __attribute__((ext_vector_type(2)));
typedef float v8f __attribute__((ext_vector_type(8)));

#define N_ROWS 8192
#define DIM    128
#define CHUNKS 8
#define CHUNK  (N_ROWS / CHUNKS)   // 1024 rows per K-chunk
#define TILES  64                  // 8x8 grid of 16x16 tiles covering 128x128

// ws layout (floats):
//   P : [2][TILES][CHUNKS][256]  partial gram tiles  (1 MB)
//   BP: [TILES]                  per-tile <Gx,Gy> partial sums
#define P_ELEMS (2 * TILES * CHUNKS * 256)
#define BP_OFF  P_ELEMS

// ---------------------------------------------------------------------------
// K1: partial Gram tiles via V_WMMA_F32_16X16X4_F32.
//   Gram G[p16+m][q16+n] = sum_r S[r][p16+m] * S[r][q16+n]
//   A(16x4): A[m][k] = S[k0+k][16p+m]   B(4x16): B[k][n] = S[k0+k][16q+n]
// Per ISA layout: lane.lo -> m (A) / n (B); (lane.hi*2 + vgpr) -> k.
// ---------------------------------------------------------------------------
__global__ __launch_bounds__(256) void gram_partial_kernel(
    const float* __restrict__ x, const float* __restrict__ y,
    float* __restrict__ ws)
{
  const int wave  = (blockIdx.x * 256 + threadIdx.x) >> 5;   // 0..1023
  const int lane  = threadIdx.x & 31;
  const int chunk = wave & (CHUNKS - 1);
  const int tile  = (wave >> 3) & (TILES - 1);
  const int mat   = wave >> 9;                               // 0 = X, 1 = Y
  const int p = tile >> 3, q = tile & 7;

  const float* __restrict__ src = (mat == 0) ? x : y;

  const int hi = lane >> 4;          // 0 or 1
  const int lo = lane & 15;
  const int cA = p * 16 + lo;        // A column in [0,128)
  const int cB = q * 16 + lo;        // B column in [0,128)

  v8f acc = {0.f, 0.f, 0.f, 0.f, 0.f, 0.f, 0.f, 0.f};

  const int k0 = chunk * CHUNK;
#pragma unroll 4
  for (int k = 0; k < CHUNK; k += 4) {
    const int rbase = (k0 + k + hi * 2) * DIM;
    v2f a, b;
    a.x = src[rbase + cA];           // row r,   col cA
    a.y = src[rbase + DIM + cA];     // row r+1, col cA
    b.x = src[rbase + cB];
    b.y = src[rbase + DIM + cB];
    // D = A x B + C, full f32. (neg_a, A, neg_b, B, c_mod, C, reuse_a, reuse_b)
    acc = __builtin_amdgcn_wmma_f32_16x16x4_f32(
        false, a, false, b, (short)0, acc, false, false);
  }

  // C/D layout: VGPR v, lanes 0-15 -> M=v, lanes 16-31 -> M=v+8; N = lane.lo
  float* out = ws + ((mat * TILES + tile) * CHUNKS + chunk) * 256;
#pragma unroll
  for (int v = 0; v < 8; ++v) {
    const int m = v + hi * 8;
    out[m * 16 + lo] = acc[v];
  }
}

// ---------------------------------------------------------------------------
// K2: per tile, sum K-chunk partials to get Gx, Gy elements; accumulate
//     sum(Gx .* Gy) per tile with a fixed-order tree reduction.
// ---------------------------------------------------------------------------
__global__ __launch_bounds__(256) void gram_combine_kernel(float* __restrict__ ws)
{
  const int tile = blockIdx.x;       // 0..63
  const int e    = threadIdx.x;      // 0..255 element within 16x16 tile
  const float* px = ws + ((0 * TILES + tile) * CHUNKS) * 256;
  const float* py = ws + ((1 * TILES + tile) * CHUNKS) * 256;

  float gx = 0.f, gy = 0.f;
#pragma unroll
  for (int c = 0; c < CHUNKS; ++c) {
    gx += px[c * 256 + e];
    gy += py[c * 256 + e];
  }

  __shared__ float red[256];
  red[e] = gx * gy;
  __syncthreads();
  for (int s = 128; s > 0; s >>= 1) {
    if (e < s) red[e] += red[e + s];
    __syncthreads();
  }
  if (e == 0) ws[BP_OFF + tile] = red[0];
}

// ---------------------------------------------------------------------------
// K3: diagonal terms + final combine.
//   S_all  = sum_{i,j} <x_i,y_j>^2  (= sum of tile partials)
//   S_d    = sum_i <x_i,y_i>
//   S_dsq  = sum_i <x_i,y_i>^2
//   loss   = (S_all - S_dsq)/(N(N-1)) - (2/N) S_d
// ---------------------------------------------------------------------------
__global__ __launch_bounds__(256) void finalize_kernel(
    const float* __restrict__ x, const float* __restrict__ y,
    const float* __restrict__ ws, float* __restrict__ out)
{
  const int t = threadIdx.x;
  float s1 = 0.f, s2 = 0.f;
  for (int i = t; i < N_ROWS; i += 256) {
    const float4* xr = (const float4*)(x + i * DIM);
    const float4* yr = (const float4*)(y + i * DIM);
    float d = 0.f;
#pragma unroll
    for (int j = 0; j < DIM / 4; ++j) {
      float4 a = xr[j], b = yr[j];
      d += a.x * b.x + a.y * b.y + a.z * b.z + a.w * b.w;
    }
    s1 += d;
    s2 += d * d;
  }
  float bp = (t < TILES) ? ws[BP_OFF + t] : 0.f;

  __shared__ float r1[256], r2[256], r3[256];
  r1[t] = s1; r2[t] = s2; r3[t] = bp;
  __syncthreads();
  for (int s = 128; s > 0; s >>= 1) {
    if (t < s) { r1[t] += r1[t + s]; r2[t] += r2[t + s]; r3[t] += r3[t + s]; }
    __syncthreads();
  }
  if (t == 0) {
    const double Nf = (double)N_ROWS;
    const double S_all = (double)r3[0];
    const double S_dsq = (double)r2[0];
    const double S_d   = (double)r1[0];
    const double loss  = (S_all - S_dsq) / (Nf * (Nf - 1.0)) - (2.0 / Nf) * S_d;
    out[0] = (float)loss;
  }
}

// ---------------------------------------------------------------------------
extern "C" void kernel_launch(void* const* d_in, const int* in_sizes, int n_in,
                              void* d_out, int out_size, void* d_ws, size_t ws_size,
                              hipStream_t stream) {
  const float* x = (const float*)d_in[0];
  const float* y = (const float*)d_in[1];
  float* ws  = (float*)d_ws;
  float* out = (float*)d_out;

  // 128 blocks x 8 waves = 1024 waves = 2 mats x 64 tiles x 8 chunks
  gram_partial_kernel<<<128, 256, 0, stream>>>(x, y, ws);
  gram_combine_kernel<<<TILES, 256, 0, stream>>>(ws);
  finalize_kernel<<<1, 256, 0, stream>>>(x, y, ws, out);
}